// PyQObservable_1322849927816
// MI455X (gfx1250) — compile-verified
//
#include <hip/hip_runtime.h>

typedef __attribute__((ext_vector_type(2))) float v2f;
typedef __attribute__((ext_vector_type(8))) float v8f;

#define BATCH 16
#define COEFF_PER_B 20

// Fixed launch geometry: 1024 blocks x 256 threads -> each thread processes
// exactly NP / (GRIDX*BLOCKX) = 8 (group, batch) pairs. Keep in sync with
// kernel_launch below.
#define GRIDX 1024u
#define BLOCKX 256u
#define TOTAL_THREADS (GRIDX * BLOCKX)          // 262144
#define NPAIR ((1u << 17) * 16u)                // 2097152
#define ITERS (NPAIR / TOTAL_THREADS)           // 8 (exact)

// ---------------------------------------------------------------------------
// 4x4 complex matmul in double precision (tiny, per-batch precompute only)
// ---------------------------------------------------------------------------
__device__ inline void cmul4(const double Ar[4][4], const double Ai[4][4],
                             const double Br[4][4], const double Bi[4][4],
                             double Cr[4][4], double Ci[4][4]) {
  for (int r = 0; r < 4; ++r)
    for (int c = 0; c < 4; ++c) {
      double sr = 0.0, si = 0.0;
      for (int k = 0; k < 4; ++k) {
        sr += Ar[r][k] * Br[k][c] - Ai[r][k] * Bi[k][c];
        si += Ar[r][k] * Bi[k][c] + Ai[r][k] * Br[k][c];
      }
      Cr[r][c] = sr;
      Ci[r][c] = si;
    }
}

// ---------------------------------------------------------------------------
// Per-batch coefficient precompute (16 threads, one per batch element):
//   M0  = acc^dag Z acc              (acc = U7@...@U0, RX chain on qubit 0)
//   M25 = Ue^dag diag(2,0,0,-2) Ue   (Ue = exp(-i t H), H = 0.5(A + A^dag))
// Stored as 20 floats per batch; also zero-initializes d_out.
// ---------------------------------------------------------------------------
__global__ void precompute_kernel(const float* __restrict__ hr,
                                  const float* __restrict__ hi,
                                  const float* __restrict__ th,
                                  const float* __restrict__ tv,
                                  float* __restrict__ coeff,
                                  float* __restrict__ out) {
  const int b = threadIdx.x;
  if (b >= BATCH) return;
  out[b] = 0.0f;

  // ---- RX chain: acc = U7 @ U6 @ ... @ U0, U = [[c,-is],[-is,c]] ----
  double ar[2][2], ai[2][2];
  {
    double hh = 0.5 * (double)th[7 * BATCH + b];
    double c = cos(hh), s = sin(hh);
    ar[0][0] = c;   ar[0][1] = 0.0; ar[1][0] = 0.0; ar[1][1] = c;
    ai[0][0] = 0.0; ai[0][1] = -s;  ai[1][0] = -s;  ai[1][1] = 0.0;
  }
  for (int i = 6; i >= 0; --i) {
    double hh = 0.5 * (double)th[i * BATCH + b];
    double c = cos(hh), s = sin(hh);
    double nr[2][2], ni[2][2];
    for (int r = 0; r < 2; ++r) {
      double a0r = ar[r][0], a0i = ai[r][0], a1r = ar[r][1], a1i = ai[r][1];
      nr[r][0] = a0r * c + s * a1i;   // row @ [[c,-is],[-is,c]]
      ni[r][0] = a0i * c - s * a1r;
      nr[r][1] = a1r * c + s * a0i;
      ni[r][1] = a1i * c - s * a0r;
    }
    for (int r = 0; r < 2; ++r)
      for (int c2 = 0; c2 < 2; ++c2) { ar[r][c2] = nr[r][c2]; ai[r][c2] = ni[r][c2]; }
  }
  // M0 = acc^dag Z acc : m_rc = conj(a0r) a0c - conj(a1r) a1c
  double m00  = (ar[0][0]*ar[0][0] + ai[0][0]*ai[0][0]) - (ar[1][0]*ar[1][0] + ai[1][0]*ai[1][0]);
  double m11  = (ar[0][1]*ar[0][1] + ai[0][1]*ai[0][1]) - (ar[1][1]*ar[1][1] + ai[1][1]*ai[1][1]);
  double m01r = (ar[0][0]*ar[0][1] + ai[0][0]*ai[0][1]) - (ar[1][0]*ar[1][1] + ai[1][0]*ai[1][1]);
  double m01i = (ar[0][0]*ai[0][1] - ai[0][0]*ar[0][1]) - (ar[1][0]*ai[1][1] - ai[1][0]*ar[1][1]);

  // ---- Ue = exp(-i t H): scaling-and-squaring + Horner Taylor (double) ----
  const double t = (double)tv[b];
  double Mr[4][4], Mi[4][4];
  for (int r = 0; r < 4; ++r)
    for (int c = 0; c < 4; ++c) {
      double hre = 0.5 * ((double)hr[(r*4 + c) * BATCH + b] + (double)hr[(c*4 + r) * BATCH + b]);
      double him = 0.5 * ((double)hi[(r*4 + c) * BATCH + b] - (double)hi[(c*4 + r) * BATCH + b]);
      Mr[r][c] =  t * him;   // -i*(x+iy) = y - i x
      Mi[r][c] = -t * hre;
    }
  double nrm = 0.0;
  for (int r = 0; r < 4; ++r) {
    double rs = 0.0;
    for (int c = 0; c < 4; ++c) rs += fabs(Mr[r][c]) + fabs(Mi[r][c]);
    nrm = fmax(nrm, rs);
  }
  int s = 0;
  while (nrm > 0.25 && s < 40) { nrm *= 0.5; ++s; }
  double sc = 1.0;
  for (int i = 0; i < s; ++i) sc *= 0.5;
  for (int r = 0; r < 4; ++r)
    for (int c = 0; c < 4; ++c) { Mr[r][c] *= sc; Mi[r][c] *= sc; }

  double Xr[4][4], Xi[4][4];
  const int NT = 12;
  for (int r = 0; r < 4; ++r)
    for (int c = 0; c < 4; ++c) {
      Xr[r][c] = (r == c ? 1.0 : 0.0) + Mr[r][c] / NT;
      Xi[r][c] = Mi[r][c] / NT;
    }
  for (int k = NT - 1; k >= 1; --k) {
    double Tr[4][4], Ti[4][4];
    cmul4(Mr, Mi, Xr, Xi, Tr, Ti);
    double inv = 1.0 / (double)k;
    for (int r = 0; r < 4; ++r)
      for (int c = 0; c < 4; ++c) {
        Xr[r][c] = (r == c ? 1.0 : 0.0) + Tr[r][c] * inv;
        Xi[r][c] = Ti[r][c] * inv;
      }
  }
  for (int i = 0; i < s; ++i) {
    double Tr[4][4], Ti[4][4];
    cmul4(Xr, Xi, Xr, Xi, Tr, Ti);
    for (int r = 0; r < 4; ++r)
      for (int c = 0; c < 4; ++c) { Xr[r][c] = Tr[r][c]; Xi[r][c] = Ti[r][c]; }
  }

  // ---- store coefficients (cross terms pre-scaled by 2 for 2*Re(m z)) ----
  float* cf = coeff + b * COEFF_PER_B;
  cf[0] = (float)m00;
  cf[1] = (float)m11;
  cf[2] = (float)(2.0 * m01r);
  cf[3] = (float)(2.0 * m01i);
  // M25_rc = 2*(conj(U[0][r]) U[0][c] - conj(U[3][r]) U[3][c])
  for (int r = 0; r < 4; ++r) {
    double d = 2.0 * ((Xr[0][r]*Xr[0][r] + Xi[0][r]*Xi[0][r]) -
                      (Xr[3][r]*Xr[3][r] + Xi[3][r]*Xi[3][r]));
    cf[4 + r] = (float)d;
  }
  const int prl[6] = {0, 0, 0, 1, 1, 2};
  const int pcl[6] = {1, 2, 3, 2, 3, 3};
  for (int k = 0; k < 6; ++k) {
    int r = prl[k], c = pcl[k];
    double mr = 2.0 * ((Xr[0][r]*Xr[0][c] + Xi[0][r]*Xi[0][c]) -
                       (Xr[3][r]*Xr[3][c] + Xi[3][r]*Xi[3][c]));
    double mi = 2.0 * ((Xr[0][r]*Xi[0][c] - Xi[0][r]*Xr[0][c]) -
                       (Xr[3][r]*Xi[3][c] - Xi[3][r]*Xr[3][c]));
    cf[8 + 2 * k] = (float)(2.0 * mr);
    cf[9 + 2 * k] = (float)(2.0 * mi);
  }
}

// ---------------------------------------------------------------------------
// One (group, batch) pair: load 8 complex amplitudes spanning index bits
// {19,17,14}, return its contribution to the expectation value.
// Plain RT loads: the 128 MB state fits in the 192 MB L2, so graph replays
// can be served at L2 bandwidth (NT hints would forfeit residency).
// ---------------------------------------------------------------------------
__device__ __forceinline__ float group_term(unsigned p, int b,
                                            const float* __restrict__ cf,
                                            const float* __restrict__ sre,
                                            const float* __restrict__ sim) {
  const unsigned g = p >> 4;
  // rest bits: g[13:0]->idx[13:0], g[15:14]->idx[16:15], g[16]->idx[18]
  const unsigned base = (g & 16383u) | (((g >> 14) & 3u) << 15) | ((g >> 16) << 18);
  const float wrest = 17.0f - 2.0f * (float)__popc((int)base);
  const unsigned a0 = (base << 4) + (unsigned)b;  // element offset idx*16 + b

  float vr[8], vi[8];
#pragma unroll
  for (int j = 0; j < 8; ++j) {  // j = b19*4 + b17*2 + b14
    const unsigned off = a0 + ((unsigned)(j & 1) << 18)         // bit14 * 16
                            + ((unsigned)((j >> 1) & 1) << 21)  // bit17 * 16
                            + ((unsigned)(j >> 2) << 23);       // bit19 * 16
    vr[j] = sre[off];
    vi[j] = sim[off];
  }

  float n2[8];
#pragma unroll
  for (int j = 0; j < 8; ++j) n2[j] = vr[j] * vr[j] + vi[j] * vi[j];

  const float slo = (n2[0] + n2[1]) + (n2[2] + n2[3]);
  const float shi = (n2[4] + n2[5]) + (n2[6] + n2[7]);
  float gp = wrest * (slo + shi) + cf[0] * slo + cf[1] * shi;

  // qubit-0 cross terms: pairs (j, j+4) over bit19
#pragma unroll
  for (int q = 0; q < 4; ++q) {
    const float zr = vr[q] * vr[q + 4] + vi[q] * vi[q + 4];
    const float zi = vr[q] * vi[q + 4] - vi[q] * vr[q + 4];
    gp += cf[2] * zr - cf[3] * zi;
  }

  // (q2,q5) quadratic form: quads o..o+3 for each bit19 value
#pragma unroll
  for (int h2 = 0; h2 < 2; ++h2) {
    const int o = h2 * 4;
    gp += cf[4] * n2[o] + cf[5] * n2[o + 1] + cf[6] * n2[o + 2] + cf[7] * n2[o + 3];
#define CROSS25(rr, cc, kk)                                                 \
    {                                                                       \
      const float zr = vr[o + rr] * vr[o + cc] + vi[o + rr] * vi[o + cc];   \
      const float zi = vr[o + rr] * vi[o + cc] - vi[o + rr] * vr[o + cc];   \
      gp += cf[8 + 2 * kk] * zr - cf[9 + 2 * kk] * zi;                      \
    }
    CROSS25(0, 1, 0)
    CROSS25(0, 2, 1)
    CROSS25(0, 3, 2)
    CROSS25(1, 2, 3)
    CROSS25(1, 3, 4)
    CROSS25(2, 3, 5)
#undef CROSS25
  }
  return gp;
}

// ---------------------------------------------------------------------------
// Single-pass streaming expectation. Thread t: batch b = t&15. A wave's 32
// lanes read 128 contiguous bytes per (j, re/im) load -> fully coalesced.
// Static trip count (8 = 4x2) with statically split even/odd accumulators:
// no per-iteration exec-mask loop checks, no cndmask accumulator select,
// and EXEC is all-ones at the WMMA by construction.
// Wave reduction: V_WMMA_F32_16X16X4_F32 with A = per-lane partials
// (row M = lane%16 == batch), B = ones -> D[m,*] = batch-m wave sum.
// ---------------------------------------------------------------------------
__global__ __launch_bounds__(BLOCKX) void expectation_kernel(
    const float* __restrict__ sre, const float* __restrict__ sim,
    const float* __restrict__ coeff, float* __restrict__ out) {
  __shared__ float sm[BATCH];
  const int tid = threadIdx.x;
  if (tid < BATCH) sm[tid] = 0.0f;
  __syncthreads();

  const unsigned t = blockIdx.x * BLOCKX + tid;
  const int b = (int)(t & 15u);

  float cf[COEFF_PER_B];
#pragma unroll
  for (int k = 0; k < COEFF_PER_B; ++k) cf[k] = coeff[b * COEFF_PER_B + k];

  float acc0 = 0.0f, acc1 = 0.0f;
#pragma unroll
  for (unsigned it = 0; it < ITERS; it += 2) {
    acc0 += group_term(t + it * TOTAL_THREADS, b, cf, sre, sim);
    acc1 += group_term(t + (it + 1) * TOTAL_THREADS, b, cf, sre, sim);
  }

  // ---- wave-level batched reduction via WMMA (EXEC uniformly all-ones) ----
  v2f av; av.x = acc0; av.y = acc1;   // A row m = {lanes m, m+16} x 2 VGPRs = 4 partials of batch m
  v2f bv; bv.x = 1.0f; bv.y = 1.0f;   // B = ones 4x16 (layout-independent)
  v8f cv = {0.f, 0.f, 0.f, 0.f, 0.f, 0.f, 0.f, 0.f};
  v8f dv = __builtin_amdgcn_wmma_f32_16x16x4_f32(false, av, false, bv,
                                                 (short)0, cv, false, false);

  const int lane = tid & 31;
  if (lane == 0) {            // D[v, 0] for v=0..7 -> batches 0..7
#pragma unroll
    for (int v = 0; v < 8; ++v) atomicAdd(&sm[v], dv[v]);
  } else if (lane == 16) {    // D[8+v, 0] for v=0..7 -> batches 8..15
#pragma unroll
    for (int v = 0; v < 8; ++v) atomicAdd(&sm[8 + v], dv[v]);
  }
  __syncthreads();
  if (tid < BATCH) atomicAdd(&out[tid], sm[tid]);
}

// ---------------------------------------------------------------------------
extern "C" void kernel_launch(void* const* d_in, const int* in_sizes, int n_in,
                              void* d_out, int out_size, void* d_ws, size_t ws_size,
                              hipStream_t stream) {
  (void)in_sizes; (void)n_in; (void)out_size; (void)ws_size;
  const float* sre = (const float*)d_in[0];
  const float* sim = (const float*)d_in[1];
  const float* hr  = (const float*)d_in[2];
  const float* hi  = (const float*)d_in[3];
  const float* th  = (const float*)d_in[4];
  const float* tv  = (const float*)d_in[5];
  float* out   = (float*)d_out;
  float* coeff = (float*)d_ws;  // 16 * 20 floats

  precompute_kernel<<<1, BATCH, 0, stream>>>(hr, hi, th, tv, coeff, out);
  expectation_kernel<<<GRIDX, BLOCKX, 0, stream>>>(sre, sim, coeff, out);
}